// TaskAttention_72859825209796
// MI455X (gfx1250) — compile-verified
//
#include <hip/hip_runtime.h>
#include <cstdint>

#define B_   4
#define T_   8
#define NF_  8192
#define C_   384
#define H_   12
#define K_   32
#define TN_  (T_ + NF_)              // 8200
#define SCALE_ 0.17677669529663687f  // 32^-0.5
#define NEG_INF (-__builtin_inff())

typedef __attribute__((ext_vector_type(2))) float v2f;
typedef __attribute__((ext_vector_type(8))) float v8f;

// ---------------------------------------------------------------------------
// K1: q[b,t,o] = sum_c x[b,t,c] * qs_w[t,c,o]
// ---------------------------------------------------------------------------
__global__ __launch_bounds__(128) void q_kernel(const float* __restrict__ x,
                                                const float* __restrict__ qs_w,
                                                float* __restrict__ qbuf) {
  __shared__ float row[C_];
  const int bid = blockIdx.x;
  const int b = bid / T_, t = bid - b * T_;
  const int tid = threadIdx.x;
  for (int c = tid; c < C_; c += 128) row[c] = x[((size_t)b * TN_ + t) * C_ + c];
  __syncthreads();
  const float* W = qs_w + (size_t)t * C_ * C_;
  for (int o = tid; o < C_; o += 128) {
    float acc = 0.f;
    for (int c = 0; c < C_; ++c) acc += row[c] * W[(size_t)c * C_ + o];
    qbuf[((size_t)b * T_ + t) * C_ + o] = acc;
  }
}

// ---------------------------------------------------------------------------
// K2: kv = feature @ kv_w  via V_WMMA_F32_16X16X4_F32
//   M = B*NF = 32768, K = 384, N = 768
//   grid (2048 M-tiles, 6 col-groups), 8 waves/WG, each wave one 16x16 C tile.
// A frag (16x4 f32): lanes 0-15 -> M=lane, regs = K{k0, k0+1};
//                    lanes 16-31 -> M=lane-16, regs = K{k0+2, k0+3}.
// B frag (4x16): mirrored (N across lanes 0-15, K striped across regs/halves).
// C tile: reg r -> M=r (lanes 0-15) / M=r+8 (lanes 16-31), N = lane&15.
// ---------------------------------------------------------------------------
__global__ __launch_bounds__(256) void kv_gemm_wmma(const float* __restrict__ x,
                                                    const float* __restrict__ kv_w,
                                                    float* __restrict__ kvbuf) {
  __shared__ float As[16 * 388];   // stride 388 (mod 64 == 4): conflict-free frag reads
  __shared__ float Bs[64 * 144];   // stride 144 (2*144 mod 64 == 32): halves disjoint
  const int tid = threadIdx.x;
  const int row0 = blockIdx.x * 16;   // global M tile start (never crosses b: NF%16==0)
  const int grp = blockIdx.y;         // 128-col group

  // Stage the full 16x384 A tile (feature rows), coalesced.
  for (int i = tid; i < 16 * C_; i += 256) {
    const int r = i / C_, c = i - r * C_;
    const int gr = row0 + r;
    const int bb = gr >> 13, nn = gr & (NF_ - 1);
    As[r * 388 + c] = x[((size_t)bb * TN_ + T_ + nn) * C_ + c];
  }

  const int lane = tid & 31;
  const int wv = tid >> 5;
  const int colLo = lane & 15;
  const int half = lane >> 4;          // 0: lanes 0-15, 1: lanes 16-31
  const int cw = wv * 16 + colLo;      // column within 128-col group
  v8f acc = {};

  for (int kc = 0; kc < 6; ++kc) {     // K chunks of 64
    __syncthreads();                   // also covers initial A-tile load
    for (int i = tid; i < 64 * 128; i += 256) {
      const int kr = i >> 7, cc = i & 127;
      Bs[kr * 144 + cc] = kv_w[(size_t)(kc * 64 + kr) * 768 + grp * 128 + cc];
    }
    __syncthreads();
#pragma unroll
    for (int kk = 0; kk < 16; ++kk) {
      const int kg = kc * 64 + kk * 4;   // global K (A staged in full)
      const int kl = kk * 4;             // chunk-local K (B)
      v2f a, bf;
      a.x  = As[colLo * 388 + kg + 2 * half];
      a.y  = As[colLo * 388 + kg + 2 * half + 1];
      bf.x = Bs[(kl + 2 * half) * 144 + cw];
      bf.y = Bs[(kl + 2 * half + 1) * 144 + cw];
      acc = __builtin_amdgcn_wmma_f32_16x16x4_f32(false, a, false, bf,
                                                  (short)0, acc, false, false);
    }
  }

  const int colG = grp * 128 + cw;
#pragma unroll
  for (int r = 0; r < 8; ++r) {
    const int row = row0 + r + half * 8;
    kvbuf[(size_t)row * 768 + colG] = acc[r];   // row == b*NF + n
  }
}

// ---------------------------------------------------------------------------
// K3: per (b,h,t): scores over NF in LDS -> iterative top-32 (tie: lowest idx,
//     matching lax.top_k) -> softmax -> attn_token + store (idx, w).
// ---------------------------------------------------------------------------
__global__ __launch_bounds__(256) void attn_topk_kernel(const float* __restrict__ qbuf,
                                                        const float* __restrict__ kvbuf,
                                                        int* __restrict__ idxbuf,
                                                        float* __restrict__ wbuf,
                                                        float* __restrict__ atok) {
  __shared__ float sc[NF_];            // 32 KB score row
  __shared__ float qh[32];
  __shared__ float rv[256];
  __shared__ int   ri[256];
  __shared__ float svals[K_];
  __shared__ int   sidx[K_];
  __shared__ float sw[K_];
  const int bid = blockIdx.x;          // == (b*H + h)*T + t
  const int b = bid / (H_ * T_);
  const int rem = bid - b * H_ * T_;
  const int h = rem / T_;
  const int t = rem - h * T_;
  const int tid = threadIdx.x;

  if (tid < 32) qh[tid] = qbuf[((size_t)b * T_ + t) * C_ + h * 32 + tid];
  __syncthreads();

  for (int n = tid; n < NF_; n += 256) {
    const float4* kp =
        reinterpret_cast<const float4*>(kvbuf + ((size_t)b * NF_ + n) * 768 + h * 32);
    float acc = 0.f;
#pragma unroll
    for (int d4 = 0; d4 < 8; ++d4) {
      const float4 kq = kp[d4];
      acc += qh[d4 * 4 + 0] * kq.x + qh[d4 * 4 + 1] * kq.y +
             qh[d4 * 4 + 2] * kq.z + qh[d4 * 4 + 3] * kq.w;
    }
    sc[n] = acc * SCALE_;
  }
  __syncthreads();

  for (int s = 0; s < K_; ++s) {
    float bv = NEG_INF; int bi = 0x7fffffff;
    for (int n = tid; n < NF_; n += 256) {
      const float v = sc[n];
      if (v > bv || (v == bv && n < bi)) { bv = v; bi = n; }
    }
    rv[tid] = bv; ri[tid] = bi;
    __syncthreads();
    for (int off = 128; off > 0; off >>= 1) {
      if (tid < off) {
        const float v2 = rv[tid + off]; const int i2 = ri[tid + off];
        if (v2 > rv[tid] || (v2 == rv[tid] && i2 < ri[tid])) { rv[tid] = v2; ri[tid] = i2; }
      }
      __syncthreads();
    }
    if (tid == 0) { svals[s] = rv[0]; sidx[s] = ri[0]; sc[ri[0]] = NEG_INF; }
    __syncthreads();
  }

  if (tid == 0) {
    const float m = svals[0];          // first pick is the max
    float ssum = 0.f;
    for (int k = 0; k < K_; ++k) { const float e = expf(svals[k] - m); sw[k] = e; ssum += e; }
    const float inv = 1.f / ssum;
    for (int k = 0; k < K_; ++k) sw[k] *= inv;
  }
  __syncthreads();

  if (tid < K_) {
    idxbuf[(size_t)bid * K_ + tid] = sidx[tid];
    wbuf[(size_t)bid * K_ + tid] = sw[tid];
  }
  if (tid < 32) {                      // attn_token slice (D=32)
    float a = 0.f;
    for (int k = 0; k < K_; ++k)
      a += sw[k] * kvbuf[((size_t)b * NF_ + sidx[k]) * 768 + 384 + h * 32 + tid];
    atok[((size_t)b * T_ + t) * C_ + h * 32 + tid] = a;
  }
}

// ---------------------------------------------------------------------------
// K5: token_out[b,t,o] = sum_c attn_token[b,t,c] * experts_w[t,c,o] -> out[:, :T]
// ---------------------------------------------------------------------------
__global__ __launch_bounds__(128) void token_out_kernel(const float* __restrict__ atok,
                                                        const float* __restrict__ experts_w,
                                                        float* __restrict__ out) {
  __shared__ float row[C_];
  const int bid = blockIdx.x;
  const int b = bid / T_, t = bid - b * T_;
  const int tid = threadIdx.x;
  for (int c = tid; c < C_; c += 128) row[c] = atok[((size_t)b * T_ + t) * C_ + c];
  __syncthreads();
  const float* W = experts_w + (size_t)t * C_ * C_;
  for (int o = tid; o < C_; o += 128) {
    float acc = 0.f;
    for (int c = 0; c < C_; ++c) acc += row[c] * W[(size_t)c * C_ + o];
    out[((size_t)b * TN_ + t) * C_ + o] = acc;
  }
}

// ---------------------------------------------------------------------------
// K6: feature_out (deterministic inverted scatter-GEMM).
//   out[b, T+n, o] = sum over entries e=(h,t,k) with idx==n of
//                    w_e * sum_d feature[b,n,h*32+d] * experts_w[t, h*32+d, o]
//   One wave per output row n; per-batch (idx,w) table in LDS; every row
//   written (zeros if no hit).
// ---------------------------------------------------------------------------
__global__ __launch_bounds__(256) void feature_out_kernel(const float* __restrict__ x,
                                                          const float* __restrict__ experts_w,
                                                          const int* __restrict__ idxbuf,
                                                          const float* __restrict__ wbuf,
                                                          float* __restrict__ out) {
  __shared__ int   sIdx[H_ * T_ * K_];   // 3072 entries, e = (h*T+t)*K + k
  __shared__ float sW[H_ * T_ * K_];
  __shared__ float sFeat[8 * C_];
  const int b = blockIdx.y;
  const int tid = threadIdx.x;
  for (int i = tid; i < H_ * T_ * K_; i += 256) {
    sIdx[i] = idxbuf[(size_t)b * (H_ * T_ * K_) + i];
    sW[i]   = wbuf[(size_t)b * (H_ * T_ * K_) + i];
  }
  const int wave = tid >> 5, lane = tid & 31;
  const int n = blockIdx.x * 8 + wave;
  const float* featRow = x + ((size_t)b * TN_ + T_ + n) * C_;
#pragma unroll
  for (int j = 0; j < 12; ++j) sFeat[wave * C_ + lane + 32 * j] = featRow[lane + 32 * j];
  __syncthreads();

  float acc[12];
#pragma unroll
  for (int j = 0; j < 12; ++j) acc[j] = 0.f;
  const float* sF = sFeat + wave * C_;

  for (int e = 0; e < H_ * T_ * K_; ++e) {   // LDS-broadcast scan
    if (sIdx[e] == n) {
      const float w = sW[e];
      const int h = e >> 8;                  // T*K = 256
      const int t = (e >> 5) & 7;            // K = 32
      const float* E = experts_w + (size_t)t * C_ * C_ + (size_t)(h * 32) * C_;
#pragma unroll
      for (int j = 0; j < 12; ++j) {
        const float* Ej = E + lane + 32 * j;
        float a = 0.f;
#pragma unroll
        for (int d = 0; d < 32; ++d) a += sF[h * 32 + d] * Ej[(size_t)d * C_];
        acc[j] += w * a;
      }
    }
  }
  float* orow = out + ((size_t)b * TN_ + T_ + n) * C_;
#pragma unroll
  for (int j = 0; j < 12; ++j) orow[lane + 32 * j] = acc[j];
}

// ---------------------------------------------------------------------------
extern "C" void kernel_launch(void* const* d_in, const int* in_sizes, int n_in,
                              void* d_out, int out_size, void* d_ws, size_t ws_size,
                              hipStream_t stream) {
  (void)in_sizes; (void)n_in; (void)out_size; (void)ws_size;
  const float* x         = (const float*)d_in[0];
  const float* qs_w      = (const float*)d_in[1];
  const float* kv_w      = (const float*)d_in[2];
  const float* experts_w = (const float*)d_in[3];
  float* out = (float*)d_out;

  float* ws     = (float*)d_ws;
  float* kvbuf  = ws;                                   // B*NF*768 f32 (~100.7 MB)
  float* qbuf   = kvbuf + (size_t)B_ * NF_ * 768;       // B*T*C
  float* atok   = qbuf + (size_t)B_ * T_ * C_;          // B*T*C
  float* wbuf   = atok + (size_t)B_ * T_ * C_;          // B*H*T*K
  int*   idxbuf = (int*)(wbuf + (size_t)B_ * H_ * T_ * K_);

  q_kernel<<<B_ * T_, 128, 0, stream>>>(x, qs_w, qbuf);
  kv_gemm_wmma<<<dim3((B_ * NF_) / 16, 6), 256, 0, stream>>>(x, kv_w, kvbuf);
  attn_topk_kernel<<<B_ * H_ * T_, 256, 0, stream>>>(qbuf, kvbuf, idxbuf, wbuf, atok);
  token_out_kernel<<<B_ * T_, 128, 0, stream>>>(atok, experts_w, out);
  feature_out_kernel<<<dim3(NF_ / 8, B_), 256, 0, stream>>>(x, experts_w, idxbuf, wbuf, out);
}